// StateRep_69810398429885
// MI455X (gfx1250) — compile-verified
//
#include <hip/hip_runtime.h>
#include <hip/hip_bf16.h>

// ---- Problem constants (from reference) ----
#define BATCH 4096
#define NBRN  4      // NB
#define SLEN  50     // S
#define KDIM  40     // P*F
#define EDIM  16     // E
#define HDIM  8      // H
#define GDIM  24     // 3*H

typedef float v2f __attribute__((ext_vector_type(2)));
typedef float v8f __attribute__((ext_vector_type(8)));

// --------------------------------------------------------------------------
// K1: fold Wp (40x16) with W_ih[c]^T (16x24) -> Wc[c] (40x32, cols 24..31 = 0)
//     bc[c][n] = b_ih[c][n] + bp . W_ih[c][n]   (padded to 32)
// --------------------------------------------------------------------------
__global__ void k_prep_weights(const float* __restrict__ Wp,
                               const float* __restrict__ bp,
                               const float* __restrict__ Wih,
                               const float* __restrict__ bih,
                               float* __restrict__ Wc,
                               float* __restrict__ bc) {
    int c = blockIdx.x;
    for (int idx = threadIdx.x; idx < KDIM * 32; idx += blockDim.x) {
        int k = idx >> 5, n = idx & 31;
        float acc = 0.0f;
        if (n < GDIM) {
            #pragma unroll
            for (int e = 0; e < EDIM; ++e)
                acc += Wp[k * EDIM + e] * Wih[(c * GDIM + n) * EDIM + e];
        }
        Wc[c * KDIM * 32 + idx] = acc;
    }
    if (threadIdx.x < 32) {
        int n = threadIdx.x;
        float acc = 0.0f;
        if (n < GDIM) {
            acc = bih[c * GDIM + n];
            #pragma unroll
            for (int e = 0; e < EDIM; ++e)
                acc += bp[e] * Wih[(c * GDIM + n) * EDIM + e];
        }
        bc[c * 32 + n] = acc;
    }
}

// --------------------------------------------------------------------------
// K2: gi[c][s][b][0..23] = prof[b][c][s][:] @ Wc[c] + bc[c]
//     V_WMMA_F32_16X16X4_F32, 10 K-chunks x 2 N-tiles per 16-row tile.
//     grid (BATCH/64, SLEN, NBRN), block 128 (4 waves, one 16-row tile each)
// --------------------------------------------------------------------------
__global__ void k_gemm_gi(const float* __restrict__ prof,
                          const float* __restrict__ Wc,
                          const float* __restrict__ bc,
                          float* __restrict__ gi) {
    const int wid  = threadIdx.x >> 5;
    const int lane = threadIdx.x & 31;
    const int r    = lane & 15;      // row-within-tile / column-within-N-tile
    const int hi   = lane >> 4;      // 0: K%4 in {0,1}, 1: K%4 in {2,3}
    const int b0   = (blockIdx.x * 4 + wid) * 16;
    const int s    = blockIdx.y;
    const int c    = blockIdx.z;

    // ---- preload B fragments for both N-tiles (Wc is 40x32, zero-padded) ----
    const float* wc = Wc + c * KDIM * 32;
    const int n0 = r, n1 = 16 + r;
    v2f bf0[10], bf1[10];
    #pragma unroll
    for (int kk = 0; kk < 10; ++kk) {
        int kb = 4 * kk + 2 * hi;
        bf0[kk] = v2f{wc[kb * 32 + n0], wc[(kb + 1) * 32 + n0]};
        bf1[kk] = v2f{wc[kb * 32 + n1], wc[(kb + 1) * 32 + n1]};
    }

    // ---- A rows: prof[(b0+r)][c][s][k] ----
    const float* arow =
        prof + ((size_t)(b0 + r) * NBRN + c) * (SLEN * KDIM) + (size_t)s * KDIM;

    v8f acc0 = {}; v8f acc1 = {};
    #pragma unroll
    for (int kk = 0; kk < 10; ++kk) {
        int kb = 4 * kk + 2 * hi;
        float2 av = *(const float2*)(arow + kb);   // 8B aligned
        v2f a = v2f{av.x, av.y};
        acc0 = __builtin_amdgcn_wmma_f32_16x16x4_f32(
            false, a, false, bf0[kk], (short)0, acc0, false, false);
        acc1 = __builtin_amdgcn_wmma_f32_16x16x4_f32(
            false, a, false, bf1[kk], (short)0, acc1, false, false);
    }

    // ---- store D + bias: VGPR q holds rows q (lanes0-15) / 8+q (lanes16-31),
    //      lane%16 = column. gi layout [c][s][b][24]. ----
    const float bias0 = bc[c * 32 + n0];
    const float bias1 = bc[c * 32 + n1];
    const size_t gbase = (size_t)(c * SLEN + s) * BATCH;
    #pragma unroll
    for (int q = 0; q < 8; ++q) {
        int row = b0 + q + 8 * hi;
        float* gp = gi + (gbase + row) * GDIM;
        gp[n0] = acc0[q] + bias0;
        if (n1 < GDIM) gp[n1] = acc1[q] + bias1;
    }
}

// --------------------------------------------------------------------------
// K3: GRU scan. One thread per (b,c); W_hh/b_hh broadcast from LDS.
//     grid (BATCH/256, NBRN), block 256
// --------------------------------------------------------------------------
__global__ void k_gru_scan(const float* __restrict__ gi,
                           const int*   __restrict__ lens,
                           const float* __restrict__ Whh,
                           const float* __restrict__ bhh,
                           float* __restrict__ hbuf) {
    __shared__ float Wl[GDIM * HDIM];
    __shared__ float bl[GDIM];
    const int c = blockIdx.y;
    for (int i = threadIdx.x; i < GDIM * HDIM; i += blockDim.x)
        Wl[i] = Whh[c * GDIM * HDIM + i];
    if (threadIdx.x < GDIM) bl[threadIdx.x] = bhh[c * GDIM + threadIdx.x];
    __syncthreads();

    const int b   = blockIdx.x * blockDim.x + threadIdx.x;
    const int len = lens[b * NBRN + c];

    float h[HDIM];
    #pragma unroll
    for (int j = 0; j < HDIM; ++j) h[j] = 0.0f;

    for (int s = 0; s < len; ++s) {
        const float* gr = gi + ((size_t)(c * SLEN + s) * BATCH + b) * GDIM;
        float g[GDIM];
        #pragma unroll
        for (int q = 0; q < 6; ++q) {              // 96B rows -> 16B aligned
            float4 v = *(const float4*)(gr + 4 * q);
            g[4 * q + 0] = v.x; g[4 * q + 1] = v.y;
            g[4 * q + 2] = v.z; g[4 * q + 3] = v.w;
        }
        float gh[GDIM];
        #pragma unroll
        for (int gq = 0; gq < GDIM; ++gq) {
            float acc = bl[gq];
            #pragma unroll
            for (int k = 0; k < HDIM; ++k) acc += h[k] * Wl[gq * HDIM + k];
            gh[gq] = acc;
        }
        #pragma unroll
        for (int j = 0; j < HDIM; ++j) {
            float rg = 1.0f / (1.0f + __expf(-(g[j] + gh[j])));
            float zg = 1.0f / (1.0f + __expf(-(g[HDIM + j] + gh[HDIM + j])));
            float ng = tanhf(g[2 * HDIM + j] + rg * gh[2 * HDIM + j]);
            h[j] = (1.0f - zg) * ng + zg * h[j];
        }
    }
    float* hp = hbuf + ((size_t)c * BATCH + b) * HDIM;
    #pragma unroll
    for (int j = 0; j < HDIM; ++j) hp[j] = h[j];
}

// --------------------------------------------------------------------------
// K4: stable descending argsort via comparison counting.
//     rank[b] = #(l[j] > l[b]) + #(j<b : l[j]==l[b]); perm[rank]=b
//     grid (4, NBRN), block 1024
// --------------------------------------------------------------------------
__global__ void k_rank_perm(const int* __restrict__ lens, int* __restrict__ perm) {
    __shared__ int l[BATCH];
    const int c = blockIdx.y;
    for (int i = threadIdx.x; i < BATCH; i += blockDim.x)
        l[i] = lens[i * NBRN + c];
    __syncthreads();
    const int b  = blockIdx.x * blockDim.x + threadIdx.x;
    const int lb = l[b];
    int rank = 0;
    for (int j = 0; j < BATCH; ++j) {
        int lj = l[j];
        rank += (int)((lj > lb) | ((lj == lb) & (j < b)));
    }
    perm[c * BATCH + rank] = b;
}

// --------------------------------------------------------------------------
// K5: out[b][c*H + j] = hbuf[c][ perm[perm[b]] ][j]
// --------------------------------------------------------------------------
__global__ void k_gather(const int* __restrict__ perm,
                         const float* __restrict__ hbuf,
                         float* __restrict__ out) {
    const int idx = blockIdx.x * blockDim.x + threadIdx.x;   // 16384
    const int b = idx >> 2, c = idx & 3;
    const int p1 = perm[c * BATCH + b];
    const int p2 = perm[c * BATCH + p1];
    const float* hp = hbuf + ((size_t)c * BATCH + p2) * HDIM;
    float* op = out + (size_t)b * (NBRN * HDIM) + c * HDIM;
    #pragma unroll
    for (int j = 0; j < HDIM; ++j) op[j] = hp[j];
}

// --------------------------------------------------------------------------
extern "C" void kernel_launch(void* const* d_in, const int* in_sizes, int n_in,
                              void* d_out, int out_size, void* d_ws, size_t ws_size,
                              hipStream_t stream) {
    const float* prof = (const float*)d_in[0];   // (B, NB, S, P, F)
    const int*   lens = (const int*)  d_in[1];   // (B, NB)
    const float* Wp   = (const float*)d_in[2];   // (40, 16)
    const float* bp   = (const float*)d_in[3];   // (16)
    const float* Wih  = (const float*)d_in[4];   // (NB, 24, 16)
    const float* Whh  = (const float*)d_in[5];   // (NB, 24, 8)
    const float* bih  = (const float*)d_in[6];   // (NB, 24)
    const float* bhh  = (const float*)d_in[7];   // (NB, 24)
    float* out = (float*)d_out;                  // (B, NB*H)

    // ---- workspace layout (bytes) ----
    char* w = (char*)d_ws;
    const size_t off_Wc   = 0;                                   // NB*40*32 f32
    const size_t off_bc   = off_Wc   + (size_t)NBRN * KDIM * 32 * 4;
    const size_t off_perm = off_bc   + (size_t)NBRN * 32 * 4;
    const size_t off_hbuf = off_perm + (size_t)NBRN * BATCH * 4;
    const size_t off_gi   = off_hbuf + (size_t)NBRN * BATCH * HDIM * 4; // 610816
    float* Wc   = (float*)(w + off_Wc);
    float* bc   = (float*)(w + off_bc);
    int*   perm = (int*)  (w + off_perm);
    float* hbuf = (float*)(w + off_hbuf);
    float* gi   = (float*)(w + off_gi);   // NB*S*B*24 f32 = 78.6 MB

    k_prep_weights<<<NBRN, 256, 0, stream>>>(Wp, bp, Wih, bih, Wc, bc);

    dim3 ggrid(BATCH / 64, SLEN, NBRN);
    k_gemm_gi<<<ggrid, 128, 0, stream>>>(prof, Wc, bc, gi);

    dim3 sgrid(BATCH / 256, NBRN);
    k_gru_scan<<<sgrid, 256, 0, stream>>>(gi, lens, Whh, bhh, hbuf);

    dim3 rgrid(4, NBRN);
    k_rank_perm<<<rgrid, 1024, 0, stream>>>(lens, perm);

    k_gather<<<64, 256, 0, stream>>>(perm, hbuf, out);
}